// GIN_28956669510067
// MI455X (gfx1250) — compile-verified
//
#include <hip/hip_runtime.h>

#define F_IN  128
#define DIM   128
#define OUTD  64
#define TILE_M 16
#define LDSS  132          // LDS row stride in floats (128+4 -> conflict-free b64 frag reads)
#define BN_EPS 1e-5f
#define SCAN_BD 256

typedef __attribute__((ext_vector_type(2))) float v2f;
typedef __attribute__((ext_vector_type(8))) float v8f;

static __device__ __forceinline__ v8f wmma4(v2f a, v2f b, v8f c) {
    // D = A(16x4 f32) * B(4x16 f32) + C(16x16 f32)
    return __builtin_amdgcn_wmma_f32_16x16x4_f32(false, a, false, b, (short)0, c, false, false);
}

static __device__ __forceinline__ void lds_fence() {
    asm volatile("s_wait_dscnt 0" ::: "memory");
}

// ===========================================================================
// CSR construction (all-integer; ~1.7M int atomics total, trivial vs 102M fp)
// ===========================================================================
__global__ __launch_bounds__(256) void k_zero_int(int* __restrict__ p, int n) {
    int i = blockIdx.x * blockDim.x + threadIdx.x;
    if (i < n) p[i] = 0;
}

__global__ __launch_bounds__(256) void k_count(const int* __restrict__ ei,
                                               int* __restrict__ deg, int E) {
    int e = blockIdx.x * blockDim.x + threadIdx.x;
    if (e < E) atomicAdd(&deg[ei[E + e]], 1);     // dst = ei[E+e]
}

// Block-level inclusive scan; block totals -> aux
__global__ __launch_bounds__(SCAN_BD) void k_scan1(const int* __restrict__ deg,
                                                   int* __restrict__ incl,
                                                   int* __restrict__ aux, int n) {
    __shared__ int sm[SCAN_BD];
    int tid = threadIdx.x;
    int gid = blockIdx.x * SCAN_BD + tid;
    int v = (gid < n) ? deg[gid] : 0;
    sm[tid] = v;
    __syncthreads();
    for (int off = 1; off < SCAN_BD; off <<= 1) {
        int t = (tid >= off) ? sm[tid - off] : 0;
        __syncthreads();
        sm[tid] += t;
        __syncthreads();
    }
    if (gid < n) incl[gid] = sm[tid];
    if (tid == SCAN_BD - 1) aux[blockIdx.x] = sm[tid];
}

// Single-block exclusive scan of the block totals (nblk <= 1024)
__global__ __launch_bounds__(1024) void k_scan2(int* __restrict__ aux, int nblk) {
    __shared__ int sm[1024];
    int tid = threadIdx.x;
    int v = (tid < nblk) ? aux[tid] : 0;
    sm[tid] = v;
    __syncthreads();
    for (int off = 1; off < 1024; off <<= 1) {
        int t = (tid >= off) ? sm[tid - off] : 0;
        __syncthreads();
        sm[tid] += t;
        __syncthreads();
    }
    if (tid < nblk) aux[tid] = sm[tid] - v;       // exclusive
}

// rowptr[i+1] = incl[i] + aux[block]; rowptr[0] = 0
__global__ __launch_bounds__(SCAN_BD) void k_scan3(const int* __restrict__ incl,
                                                   const int* __restrict__ aux,
                                                   int* __restrict__ rowptr,
                                                   int* __restrict__ cur, int n) {
    int gid = blockIdx.x * SCAN_BD + threadIdx.x;
    if (gid < n) {
        int r = incl[gid] + aux[blockIdx.x];
        rowptr[gid + 1] = r;
    }
    if (gid == 0) rowptr[0] = 0;
}

__global__ __launch_bounds__(256) void k_copy_int(const int* __restrict__ src,
                                                  int* __restrict__ dst, int n) {
    int i = blockIdx.x * blockDim.x + threadIdx.x;
    if (i < n) dst[i] = src[i];
}

__global__ __launch_bounds__(256) void k_fill(const int* __restrict__ ei,
                                              int* __restrict__ cur,
                                              int* __restrict__ col, int E) {
    int e = blockIdx.x * blockDim.x + threadIdx.x;
    if (e < E) {
        int d = ei[E + e];
        int pos = atomicAdd(&cur[d], 1);
        col[pos] = ei[e];                         // src
    }
}

// ===========================================================================
// GIN aggregation as a gather: agg[i] = feat[i] + sum_{j in N(i)} feat[j]
// One wave per node; each lane owns a float4 slice (32*4 = 128 features).
// ===========================================================================
__global__ __launch_bounds__(256) void k_gather(const float* __restrict__ feat,
                                                const int* __restrict__ rowptr,
                                                const int* __restrict__ col,
                                                float* __restrict__ agg, int nNodes) {
    int node = blockIdx.x * (blockDim.x >> 5) + (threadIdx.x >> 5);
    int lane = threadIdx.x & 31;
    if (node >= nNodes) return;                   // wave-uniform
    int beg = rowptr[node];
    int end = rowptr[node + 1];
    float4 acc = ((const float4*)(feat + (size_t)node * DIM))[lane];  // self term
    for (int e = beg; e < end; ++e) {
        int s = col[e];
        float4 v = ((const float4*)(feat + (size_t)s * DIM))[lane];
        acc.x += v.x; acc.y += v.y; acc.z += v.z; acc.w += v.w;
    }
    ((float4*)(agg + (size_t)node * DIM))[lane] = acc;
}

// ===========================================================================
// Shared WMMA helpers
// ===========================================================================
static __device__ __forceinline__ void load_tile(const float* __restrict__ g,
                                                 float* tl, int lane) {
    const float4* s4 = (const float4*)g;
    #pragma unroll
    for (int r = 0; r < TILE_M; ++r) {
        float4 v = s4[r * (DIM / 4) + lane];
        float* d = tl + r * LDSS + lane * 4;
        d[0] = v.x; d[1] = v.y; d[2] = v.z; d[3] = v.w;
    }
}

template <int NT>
static __device__ __forceinline__ void mm_tiles(const float* tl,
                                                const float* __restrict__ W, int ldw,
                                                int lane, v8f acc[NT]) {
    const int ar = lane & 15;              // A: M row
    const int ko = (lane >> 4) << 1;       // A/B: K sub-offset (0 or 2)
    const int bc = lane & 15;              // B: N col within tile
    for (int kk = 0; kk < DIM / 4; ++kk) {
        const int k0 = kk * 4;
        const float* ap = tl + ar * LDSS + k0 + ko;
        v2f a; a.x = ap[0]; a.y = ap[1];
        const float* bp = W + (size_t)(k0 + ko) * ldw + bc;
        #pragma unroll
        for (int t = 0; t < NT; ++t) {
            v2f b; b.x = bp[t * 16]; b.y = bp[ldw + t * 16];
            acc[t] = wmma4(a, b, acc[t]);
        }
    }
}

template <int NT>
static __device__ __forceinline__ void zero_acc(v8f acc[NT]) {
    #pragma unroll
    for (int t = 0; t < NT; ++t)
        #pragma unroll
        for (int v = 0; v < 8; ++v) acc[t][v] = 0.0f;
}

// ===========================================================================
// MLP1: h1 = ReLU(BN1(ReLU(agg@W1a+b1a)@W1b+b1b))
// ===========================================================================
__global__ __launch_bounds__(128) void gin_mlp1(
    const float* __restrict__ agg,
    const float* __restrict__ W1a, const float* __restrict__ b1a,
    const float* __restrict__ W1b, const float* __restrict__ b1b,
    const float* __restrict__ g1, const float* __restrict__ be1,
    const float* __restrict__ m1, const float* __restrict__ v1,
    float* __restrict__ h1, int nTiles) {
    __shared__ float lds[4][TILE_M * LDSS];
    const int wave = threadIdx.x >> 5;
    const int lane = threadIdx.x & 31;
    const int tile = blockIdx.x * 4 + wave;
    if (tile >= nTiles) return;          // wave-uniform: EXEC stays all-ones
    float* tl = lds[wave];
    const int rowBase = tile * TILE_M;

    load_tile(agg + (size_t)rowBase * DIM, tl, lane);
    lds_fence();

    v8f acc[8];
    zero_acc<8>(acc);
    mm_tiles<8>(tl, W1a, DIM, lane, acc);

    const int cr = (lane >> 4) * 8;      // C/D: row offset (0 or 8)
    const int cc = lane & 15;            // C/D: col within tile
    lds_fence();
    #pragma unroll
    for (int t = 0; t < 8; ++t) {
        const float bs = b1a[t * 16 + cc];
        #pragma unroll
        for (int v = 0; v < 8; ++v)
            tl[(cr + v) * LDSS + t * 16 + cc] = fmaxf(acc[t][v] + bs, 0.0f);
    }
    lds_fence();

    zero_acc<8>(acc);
    mm_tiles<8>(tl, W1b, DIM, lane, acc);

    #pragma unroll
    for (int t = 0; t < 8; ++t) {
        const int col = t * 16 + cc;
        const float sc = g1[col] * rsqrtf(v1[col] + BN_EPS);
        const float sh = be1[col] - m1[col] * sc;
        const float bb = b1b[col];
        #pragma unroll
        for (int v = 0; v < 8; ++v) {
            float val = fmaxf((acc[t][v] + bb) * sc + sh, 0.0f);
            h1[(size_t)(rowBase + cr + v) * DIM + col] = val;
        }
    }
}

// ===========================================================================
// MLP2 + BN2 + final projection: out = BN2(ReLU(agg@W2a+b2a)@W2b+b2b) @ Wf + bf
// ===========================================================================
__global__ __launch_bounds__(128) void gin_mlp2(
    const float* __restrict__ agg,
    const float* __restrict__ W2a, const float* __restrict__ b2a,
    const float* __restrict__ W2b, const float* __restrict__ b2b,
    const float* __restrict__ g2, const float* __restrict__ be2,
    const float* __restrict__ m2, const float* __restrict__ v2,
    const float* __restrict__ Wf, const float* __restrict__ bf,
    float* __restrict__ out, int nTiles) {
    __shared__ float lds[4][TILE_M * LDSS];
    const int wave = threadIdx.x >> 5;
    const int lane = threadIdx.x & 31;
    const int tile = blockIdx.x * 4 + wave;
    if (tile >= nTiles) return;
    float* tl = lds[wave];
    const int rowBase = tile * TILE_M;

    load_tile(agg + (size_t)rowBase * DIM, tl, lane);
    lds_fence();

    v8f acc[8];
    zero_acc<8>(acc);
    mm_tiles<8>(tl, W2a, DIM, lane, acc);

    const int cr = (lane >> 4) * 8;
    const int cc = lane & 15;
    lds_fence();
    #pragma unroll
    for (int t = 0; t < 8; ++t) {
        const float bs = b2a[t * 16 + cc];
        #pragma unroll
        for (int v = 0; v < 8; ++v)
            tl[(cr + v) * LDSS + t * 16 + cc] = fmaxf(acc[t][v] + bs, 0.0f);
    }
    lds_fence();

    zero_acc<8>(acc);
    mm_tiles<8>(tl, W2b, DIM, lane, acc);

    // BN2 (no ReLU) -> back to LDS for the final projection
    lds_fence();
    #pragma unroll
    for (int t = 0; t < 8; ++t) {
        const int col = t * 16 + cc;
        const float sc = g2[col] * rsqrtf(v2[col] + BN_EPS);
        const float sh = be2[col] - m2[col] * sc;
        const float bb = b2b[col];
        #pragma unroll
        for (int v = 0; v < 8; ++v)
            tl[(cr + v) * LDSS + col] = (acc[t][v] + bb) * sc + sh;
    }
    lds_fence();

    v8f accf[4];
    zero_acc<4>(accf);
    mm_tiles<4>(tl, Wf, OUTD, lane, accf);

    #pragma unroll
    for (int t = 0; t < 4; ++t) {
        const int col = t * 16 + cc;
        const float bv = bf[col];
        #pragma unroll
        for (int v = 0; v < 8; ++v)
            out[(size_t)(rowBase + cr + v) * OUTD + col] = accf[t][v] + bv;
    }
}

// ===========================================================================
extern "C" void kernel_launch(void* const* d_in, const int* in_sizes, int n_in,
                              void* d_out, int out_size, void* d_ws, size_t ws_size,
                              hipStream_t stream) {
    const float* x   = (const float*)d_in[0];
    const int*   ei  = (const int*)d_in[1];
    const float* W1a = (const float*)d_in[2];
    const float* b1a = (const float*)d_in[3];
    const float* W1b = (const float*)d_in[4];
    const float* b1b = (const float*)d_in[5];
    const float* g1  = (const float*)d_in[6];
    const float* be1 = (const float*)d_in[7];
    const float* m1  = (const float*)d_in[8];
    const float* v1  = (const float*)d_in[9];
    const float* W2a = (const float*)d_in[10];
    const float* b2a = (const float*)d_in[11];
    const float* W2b = (const float*)d_in[12];
    const float* b2b = (const float*)d_in[13];
    const float* g2  = (const float*)d_in[14];
    const float* be2 = (const float*)d_in[15];
    const float* m2  = (const float*)d_in[16];
    const float* v2  = (const float*)d_in[17];
    const float* Wf  = (const float*)d_in[18];
    const float* bf  = (const float*)d_in[19];
    float* out = (float*)d_out;

    const int N = in_sizes[0] / F_IN;
    const int E = in_sizes[1] / 2;
    const int nblk = (N + SCAN_BD - 1) / SCAN_BD;   // scan blocks (<= 1024)

    // Workspace layout
    float* bufA  = (float*)d_ws;                 // agg (both layers)
    float* bufB  = bufA + (size_t)N * DIM;       // h1
    int* rowptr  = (int*)(bufB + (size_t)N * DIM);  // N+1
    int* deg     = rowptr + (N + 1);             // N
    int* incl    = deg + N;                      // N
    int* cur     = incl + N;                     // N
    int* col     = cur + N;                      // E
    int* aux     = col + E;                      // nblk

    // ---- CSR build (edge structure is static within a launch) ----
    k_zero_int<<<(N + 255) / 256, 256, 0, stream>>>(deg, N);
    k_count<<<(E + 255) / 256, 256, 0, stream>>>(ei, deg, E);
    k_scan1<<<nblk, SCAN_BD, 0, stream>>>(deg, incl, aux, N);
    k_scan2<<<1, 1024, 0, stream>>>(aux, nblk);
    k_scan3<<<nblk, SCAN_BD, 0, stream>>>(incl, aux, rowptr, cur, N);
    k_copy_int<<<(N + 255) / 256, 256, 0, stream>>>(rowptr, cur, N);
    k_fill<<<(E + 255) / 256, 256, 0, stream>>>(ei, cur, col, E);

    const int nTiles = (N + TILE_M - 1) / TILE_M;
    const int gatherBlocks = (N + 7) / 8;        // 8 waves (nodes) per 256-thread block

    // ---- Layer 1 ----
    k_gather<<<gatherBlocks, 256, 0, stream>>>(x, rowptr, col, bufA, N);
    gin_mlp1<<<(nTiles + 3) / 4, 128, 0, stream>>>(bufA, W1a, b1a, W1b, b1b,
                                                   g1, be1, m1, v1, bufB, nTiles);
    // ---- Layer 2 + projection ----
    k_gather<<<gatherBlocks, 256, 0, stream>>>(bufB, rowptr, col, bufA, N);
    gin_mlp2<<<(nTiles + 3) / 4, 128, 0, stream>>>(bufA, W2a, b2a, W2b, b2b,
                                                   g2, be2, m2, v2, Wf, bf, out, nTiles);
}